// ModularDecoder_71502615544596
// MI455X (gfx1250) — compile-verified
//
#include <hip/hip_runtime.h>
#include <hip/hip_bf16.h>
#include <math.h>

#define BATCH   8192
#define LATENT  128
#define HID     400
#define OUTN    4096
#define NEXP    16
#define PADR    64                  // expert-group padding (row-tile size)
#define RT      144                 // max 64-row tiles: 8192/64 + 16
#define PERM_N  (RT * PADR)         // 9216

typedef float v2f __attribute__((ext_vector_type(2)));
typedef float v8f __attribute__((ext_vector_type(8)));

static __device__ __forceinline__ v8f wmma4(v2f a, v2f b, v8f c) {
    // V_WMMA_F32_16X16X4_F32 : D(16x16,f32) = A(16x4) * B(4x16) + C
    return __builtin_amdgcn_wmma_f32_16x16x4_f32(
        /*neg_a=*/false, a, /*neg_b=*/false, b,
        /*c_mod=*/(short)0, c, /*reuse_a=*/false, /*reuse_b=*/false);
}

// ---------------- workspace layout (int32 units) ----------------
// [0..15]   counts
// [16..31]  cursor
// [32..48]  padded base offsets (base[16] = padded total)
// [64..64+PERM_N)  perm (sample index per padded row, -1 = pad)
// floats at int offset 64+PERM_N : H [PERM_N x HID]

__global__ void k_init(int* wsI) {
    int i = blockIdx.x * blockDim.x + threadIdx.x;
    if (i < 32) wsI[i] = 0;                 // counts + cursor
    if (i < PERM_N) wsI[64 + i] = -1;       // perm
}

__global__ void k_count(const int* __restrict__ aidx, int* wsI) {
    int i = blockIdx.x * blockDim.x + threadIdx.x;
    if (i < BATCH) atomicAdd(&wsI[aidx[i]], 1);
}

__global__ void k_scan(int* wsI) {
    if (threadIdx.x == 0 && blockIdx.x == 0) {
        int pb = 0;
        for (int e = 0; e < NEXP; ++e) {
            wsI[32 + e] = pb;
            pb += ((wsI[e] + (PADR - 1)) / PADR) * PADR;
        }
        wsI[32 + NEXP] = pb;
    }
}

__global__ void k_scatter(const int* __restrict__ aidx, int* wsI) {
    int i = blockIdx.x * blockDim.x + threadIdx.x;
    if (i < BATCH) {
        int e = aidx[i];
        int p = atomicAdd(&wsI[16 + e], 1);
        wsI[64 + wsI[32 + e] + p] = i;
    }
}

// ---------------- layer 1: H = relu(z @ W1[e]^T + b1[e]), permuted rows ----
// grid (RT, HID/16), block 32 (1 wave). Wave = 64 rows x 16 hid cols.
__global__ void k_layer1(const float* __restrict__ z, const float* __restrict__ W1,
                         const float* __restrict__ b1, const int* __restrict__ aidx,
                         const int* __restrict__ perm, float* __restrict__ H) {
    const int r    = blockIdx.x;
    const int c    = blockIdx.y;          // hid col tile
    const int lane = threadIdx.x & 31;
    const int n    = lane & 15;
    const int kh   = lane >> 4;
    const int s0   = perm[r * PADR];
    if (s0 < 0) return;                    // wave-uniform: EXEC stays all-ones
    const int e = aidx[s0];

    int samp[4];
#pragma unroll
    for (int t = 0; t < 4; ++t) {
        int s = perm[r * PADR + t * 16 + n];
        samp[t] = (s < 0) ? 0 : s;         // pad rows compute garbage, masked later
    }
    const float* bptr = W1 + (size_t)e * HID * LATENT
                           + (size_t)(c * 16 + n) * LATENT + 2 * kh;
    const float* ap[4];
#pragma unroll
    for (int t = 0; t < 4; ++t) ap[t] = z + (size_t)samp[t] * LATENT + 2 * kh;

    v8f acc[4] = {};
#pragma unroll 4
    for (int kb = 0; kb < LATENT; kb += 4) {
        v2f b = *(const v2f*)(bptr + kb);
#pragma unroll
        for (int t = 0; t < 4; ++t) {
            v2f a = *(const v2f*)(ap[t] + kb);
            acc[t] = wmma4(a, b, acc[t]);
        }
    }
    const float bias = b1[e * HID + c * 16 + n];
#pragma unroll
    for (int t = 0; t < 4; ++t) {
#pragma unroll
        for (int v = 0; v < 8; ++v) {
            int pr = r * PADR + t * 16 + v + 8 * kh;
            float h = acc[t][v] + bias;
            H[(size_t)pr * HID + c * 16 + n] = fmaxf(h, 0.0f);
        }
    }
}

// ---------------- layer 2: out[s] = sigmoid(H @ W2[e]^T + b2[e]) -----------
// grid (RT, 8), block 256 (8 waves). Wave = 64 rows x 64 out cols (4x4 frags).
__global__ void k_layer2(const float* __restrict__ H, const float* __restrict__ W2,
                         const float* __restrict__ b2, const int* __restrict__ aidx,
                         const int* __restrict__ perm, float* __restrict__ out) {
    const int r    = blockIdx.x;
    const int wave = threadIdx.x >> 5;
    const int lane = threadIdx.x & 31;
    const int cb   = (blockIdx.y * 8 + wave) * 64;   // column base of 64-wide tile
    const int n    = lane & 15;
    const int kh   = lane >> 4;
    const int s0   = perm[r * PADR];
    if (s0 < 0) return;                    // wave-uniform
    const int e = aidx[s0];

    const float* ap[4];
#pragma unroll
    for (int t = 0; t < 4; ++t)
        ap[t] = H + (size_t)(r * PADR + t * 16 + n) * HID + 2 * kh;
    const float* bp[4];
#pragma unroll
    for (int u = 0; u < 4; ++u)
        bp[u] = W2 + (size_t)e * OUTN * HID + (size_t)(cb + u * 16 + n) * HID + 2 * kh;

    v8f acc[4][4] = {};
    for (int kb = 0; kb < HID; kb += 4) {
        v2f a[4], b[4];
#pragma unroll
        for (int t = 0; t < 4; ++t) a[t] = *(const v2f*)(ap[t] + kb);
#pragma unroll
        for (int u = 0; u < 4; ++u) b[u] = *(const v2f*)(bp[u] + kb);
#pragma unroll
        for (int t = 0; t < 4; ++t)
#pragma unroll
            for (int u = 0; u < 4; ++u)
                acc[t][u] = wmma4(a[t], b[u], acc[t][u]);
    }

#pragma unroll
    for (int t = 0; t < 4; ++t) {
#pragma unroll
        for (int v = 0; v < 8; ++v) {
            int pr = r * PADR + t * 16 + v + 8 * kh;
            int s  = perm[pr];
            if (s >= 0) {
#pragma unroll
                for (int u = 0; u < 4; ++u) {
                    int col = cb + u * 16 + n;
                    float x = acc[t][u][v] + b2[e * OUTN + col];
                    out[(size_t)s * OUTN + col] = 1.0f / (1.0f + __expf(-x));
                }
            }
        }
    }
}

extern "C" void kernel_launch(void* const* d_in, const int* in_sizes, int n_in,
                              void* d_out, int out_size, void* d_ws, size_t ws_size,
                              hipStream_t stream) {
    const float* z    = (const float*)d_in[0];
    const float* W1   = (const float*)d_in[1];
    const float* b1   = (const float*)d_in[2];
    const float* W2   = (const float*)d_in[3];
    const float* b2   = (const float*)d_in[4];
    const int*   aidx = (const int*)d_in[5];
    float* out = (float*)d_out;

    int*   wsI  = (int*)d_ws;
    int*   perm = wsI + 64;
    float* H    = (float*)(wsI + 64 + PERM_N);

    k_init<<<(PERM_N + 255) / 256, 256, 0, stream>>>(wsI);
    k_count<<<(BATCH + 255) / 256, 256, 0, stream>>>(aidx, wsI);
    k_scan<<<1, 32, 0, stream>>>(wsI);
    k_scatter<<<(BATCH + 255) / 256, 256, 0, stream>>>(aidx, wsI);

    dim3 g1(RT, HID / 16);
    k_layer1<<<g1, 32, 0, stream>>>(z, W1, b1, aidx, perm, H);

    dim3 g2(RT, 8);
    k_layer2<<<g2, 256, 0, stream>>>(H, W2, b2, aidx, perm, out);
}